// Decoder_51178830299763
// MI455X (gfx1250) — compile-verified
//
#include <hip/hip_runtime.h>
#include <stdint.h>

// Problem constants (match the reference).
#define B_ 512
#define S_ 2048
#define T_ 48
#define CH 64            // time-steps per emission chunk (12 KB per buffer)
#define NCH (S_ / CH)    // 32 chunks
#define CK 128           // backtrack chunk (rows staged in LDS)
#define NCB (S_ / CK)    // 16 backtrack chunks

// d_ws layout:
//   [0, BPBYTES)            : backpointers uint8 [B][S][T] (row t=0 unused)
//   [BPBYTES, BPBYTES+4*B)  : last_tag int32 per batch
#define BPBYTES ((size_t)B_ * (size_t)S_ * (size_t)T_)

// ---- CDNA5 async global->LDS helpers (gfx1250) --------------------------
// Low 32 bits of a flat pointer into __shared__ are the LDS byte address
// (the LDS aperture keeps the offset in addr[31:0]).
__device__ __forceinline__ unsigned lds_u32(const void* p) {
  return (unsigned)(uintptr_t)p;
}

// GVS mode: mem = SADDR(64b sgpr pair) + VADDR(32b byte offset); dest LDS
// address comes from the first VGPR operand. Tracked by ASYNCcnt.
__device__ __forceinline__ void async_load_b128(unsigned lds_addr,
                                                unsigned voff,
                                                const void* sbase) {
  asm volatile("global_load_async_to_lds_b128 %0, %1, %2 offset:0"
               :
               : "v"(lds_addr), "v"(voff), "s"(sbase)
               : "memory");
}

// Streaming variant: non-temporal load hint so the once-read emission
// stream does not evict the backpointers from the 192 MB L2.
__device__ __forceinline__ void async_load_b128_nt(unsigned lds_addr,
                                                   unsigned voff,
                                                   const void* sbase) {
  asm volatile("global_load_async_to_lds_b128 %0, %1, %2 offset:0 th:TH_LOAD_NT"
               :
               : "v"(lds_addr), "v"(voff), "s"(sbase)
               : "memory");
}

__device__ __forceinline__ void wait_asynccnt0() {
  asm volatile("s_wait_asynccnt 0" ::: "memory");
}

// ---- Forward Viterbi ----------------------------------------------------
__global__ __launch_bounds__(64) void viterbi_forward(
    const float* __restrict__ feats, const float* __restrict__ start_tr,
    const float* __restrict__ end_tr, const float* __restrict__ trans,
    float* __restrict__ out_score, unsigned char* __restrict__ bp,
    int* __restrict__ last_tag) {
  __shared__ __align__(16) float s_emit[2][CH * T_];  // 2 x 12 KB
  __shared__ __align__(16) float s_score[2][64];      // ping-pong scores

  const int tid = threadIdx.x;
  const int b = blockIdx.x;
  const float* fb = feats + (size_t)b * S_ * T_;

  // Kick off DMA of emission chunk 0 into buffer 0 (12 async b128 per wave).
  {
    unsigned lb = lds_u32(&s_emit[0][0]);
    #pragma unroll
    for (int i = 0; i < 12; ++i) {
      unsigned off = (unsigned)((i * 64 + tid) * 16);
      async_load_b128_nt(lb + off, off, fb);
    }
  }

  // Cache this lane's transition column trans[prev][cur=tid] in VGPRs and
  // initialize scores from start_transitions + emissions[t=0] (overlaps DMA).
  float tcol[T_];
  if (tid < T_) {
    #pragma unroll
    for (int p = 0; p < T_; ++p) tcol[p] = trans[p * T_ + tid];
    s_score[0][tid] = start_tr[tid] + fb[tid];
  }

  wait_asynccnt0();
  __syncthreads();

  // Prefetch chunk 1 into buffer 1; it lands while we process chunk 0.
  {
    unsigned lb = lds_u32(&s_emit[1][0]);
    unsigned mb = (unsigned)(CH * T_ * 4);
    #pragma unroll
    for (int i = 0; i < 12; ++i) {
      unsigned off = (unsigned)((i * 64 + tid) * 16);
      async_load_b128_nt(lb + off, mb + off, fb);
    }
  }

  int pbuf = 0;
  for (int c = 0; c < NCH; ++c) {
    const int ebuf = c & 1;
    const int t0 = c * CH;
    const int tb = (c == 0) ? 1 : 0;  // t=0 consumed by the init
    for (int tl = tb; tl < CH; ++tl) {
      const int t = t0 + tl;
      if (tid < T_) {
        const float4* sv4 = (const float4*)&s_score[pbuf][0];  // LDS bcast
        float m = -3.0e38f;
        int mi = 0;
        #pragma unroll
        for (int q = 0; q < T_ / 4; ++q) {
          float4 s4 = sv4[q];
          float c0 = s4.x + tcol[q * 4 + 0];
          float c1 = s4.y + tcol[q * 4 + 1];
          float c2 = s4.z + tcol[q * 4 + 2];
          float c3 = s4.w + tcol[q * 4 + 3];
          // strict '>' keeps the FIRST max index (matches jnp.argmax ties)
          if (c0 > m) { m = c0; mi = q * 4 + 0; }
          if (c1 > m) { m = c1; mi = q * 4 + 1; }
          if (c2 > m) { m = c2; mi = q * 4 + 2; }
          if (c3 > m) { m = c3; mi = q * 4 + 3; }
        }
        bp[((size_t)b * S_ + t) * T_ + tid] = (unsigned char)mi;
        s_score[pbuf ^ 1][tid] = m + s_emit[ebuf][tl * T_ + tid];
      }
      pbuf ^= 1;
      __syncthreads();  // one barrier/step thanks to score ping-pong
    }
    // Chunk boundary: chunk c+1's DMA must have landed before we read it,
    // and everyone must be done reading ebuf before chunk c+2 overwrites it.
    wait_asynccnt0();
    __syncthreads();
    if (c + 2 < NCH) {
      unsigned lb = lds_u32(&s_emit[ebuf][0]);
      unsigned mb = (unsigned)((c + 2) * CH * T_ * 4);
      #pragma unroll
      for (int i = 0; i < 12; ++i) {
        unsigned off = (unsigned)((i * 64 + tid) * 16);
        async_load_b128_nt(lb + off, mb + off, fb);
      }
    }
  }

  // Final: add end_transitions, reduce to best score + last tag.
  if (tid < T_) s_score[pbuf][tid] += end_tr[tid];
  __syncthreads();
  if (tid == 0) {
    float m = -3.0e38f;
    int mi = 0;
    for (int k = 0; k < T_; ++k) {
      float v = s_score[pbuf][k];
      if (v > m) { m = v; mi = k; }
    }
    out_score[b] = m;
    last_tag[b] = mi;
  }
}

// ---- Backtrack ----------------------------------------------------------
// One block per batch. Backpointer rows are DMA'd into a ping-pong LDS
// buffer with async-to-LDS loads (L2-resident: 50 MB of bp < 192 MB L2);
// the DMA for chunk c-1 overlaps lane 0's serial chain walk of chunk c.
__global__ __launch_bounds__(64) void viterbi_backtrack(
    const unsigned char* __restrict__ bp, const int* __restrict__ last_tag,
    float* __restrict__ tags_out) {
  __shared__ __align__(16) unsigned char rows[2][CK * T_];  // 2 x 6 KB
  __shared__ int tag_chunk[CK];

  const int tid = threadIdx.x;
  const int b = blockIdx.x;
  const unsigned char* bpb = bp + (size_t)b * S_ * T_;
  float* tout = tags_out + (size_t)b * S_;

  // Stage the top chunk (c = NCB-1, 127 rows) into buffer 0.
  {
    const int t_lo = (NCB - 1) * CK;
    const int n16 = ((S_ - 1 - t_lo) * T_) / 16;  // 127*48/16 = 381
    unsigned lb = lds_u32(&rows[0][0]);
    unsigned mb = (unsigned)((t_lo + 1) * T_);
    #pragma unroll
    for (int i = 0; i < 6; ++i) {
      int j = i * 64 + tid;
      if (j < n16)
        async_load_b128(lb + (unsigned)(j * 16), mb + (unsigned)(j * 16), bpb);
    }
  }

  int carry = last_tag[b];  // only lane 0's copy is used
  if (tid == 0) tout[S_ - 1] = (float)carry;

  wait_asynccnt0();
  __syncthreads();

  for (int c = NCB - 1; c >= 0; --c) {
    const int buf = (NCB - 1 - c) & 1;
    const int t_lo = c * CK;
    const int t_hi = min((c + 1) * CK, S_ - 1) - 1;  // inclusive
    // Prefetch chunk c-1 (full 128 rows = 384 x b128) into the other
    // buffer; the DMA overlaps the serial walk below.
    if (c > 0) {
      unsigned lb = lds_u32(&rows[buf ^ 1][0]);
      unsigned mb = (unsigned)(((c - 1) * CK + 1) * T_);
      #pragma unroll
      for (int i = 0; i < 6; ++i) {
        unsigned off = (unsigned)((i * 64 + tid) * 16);
        async_load_b128(lb + off, mb + off, bpb);
      }
    }
    // Serial chain walk through LDS (bp row t+1 decides tag at t).
    if (tid == 0) {
      int tag = carry;
      for (int t = t_hi; t >= t_lo; --t) {
        tag = (int)rows[buf][(t - t_lo) * T_ + tag];
        tag_chunk[t - t_lo] = tag;
      }
      carry = tag;
    }
    __syncthreads();
    const int nrows = t_hi - t_lo + 1;
    for (int k = tid; k < nrows; k += 64) tout[t_lo + k] = (float)tag_chunk[k];
    // Next chunk's DMA must have landed (and tag_chunk consumed) before
    // the next iteration walks buf^1 / overwrites tag_chunk.
    wait_asynccnt0();
    __syncthreads();
  }
}

// ---- Launch -------------------------------------------------------------
extern "C" void kernel_launch(void* const* d_in, const int* in_sizes, int n_in,
                              void* d_out, int out_size, void* d_ws,
                              size_t ws_size, hipStream_t stream) {
  const float* feats = (const float*)d_in[0];
  const float* start_tr = (const float*)d_in[1];
  const float* end_tr = (const float*)d_in[2];
  const float* trans = (const float*)d_in[3];

  float* out = (float*)d_out;
  float* out_score = out;      // [512] best path scores
  float* out_tags = out + B_;  // [512*2048] tags (exact small ints as f32)

  unsigned char* bp = (unsigned char*)d_ws;
  int* last_tag = (int*)((char*)d_ws + BPBYTES);

  viterbi_forward<<<B_, 64, 0, stream>>>(feats, start_tr, end_tr, trans,
                                         out_score, bp, last_tag);
  viterbi_backtrack<<<B_, 64, 0, stream>>>(bp, last_tag, out_tags);
}